// GeneralNetworkedODE_79053168050862
// MI455X (gfx1250) — compile-verified
//
#include <hip/hip_runtime.h>
#include <hip/hip_bf16.h>

typedef __attribute__((ext_vector_type(2))) float v2f;
typedef __attribute__((ext_vector_type(8))) float v8f;

#define NAG 64
#define HID 8
#define WAVES_PER_BLOCK 8
#define BLOCK 256

__device__ __forceinline__ float fast_tanhf(float x) {
#if __has_builtin(__builtin_amdgcn_tanhf)
  return __builtin_amdgcn_tanhf(x);          // v_tanh_f32 (gfx1250 TRANS op)
#else
  float r;
  asm("v_tanh_f32 %0, %1" : "=v"(r) : "v"(x));
  return r;
#endif
}

__global__ __launch_bounds__(BLOCK) void netode_kernel(
    const float* __restrict__ x,   const float* __restrict__ W1,
    const float* __restrict__ b1,  const float* __restrict__ W2,
    const float* __restrict__ b2,  const float* __restrict__ Wc1,
    const float* __restrict__ bc1, const float* __restrict__ Wc2,
    const float* __restrict__ bc2, float* __restrict__ out,
    int Bsz, int waves_total)
{
  // per-wave private LDS slices (DS ops are in-order within a wave)
  __shared__ float lds_x[WAVES_PER_BLOCK][NAG];
  __shared__ float lds_c[WAVES_PER_BLOCK][NAG + 16];  // +16 scrap slots for lanes 16..31

  const int lane = threadIdx.x & 31;
  const int wib  = threadIdx.x >> 5;
  // wave id is uniform across the wave by construction; make it an SGPR so the
  // batch loop compiles as a scalar s_cbranch loop (no EXEC-mask bookkeeping,
  // EXEC provably all-ones at every WMMA).
  const int gw = __builtin_amdgcn_readfirstlane(blockIdx.x * WAVES_PER_BLOCK + wib);

  const int n0 = lane * 2, n1 = n0 + 1;                // agents owned by this lane

  float* const lx = &lds_x[wib][0];
  float* const lc = &lds_c[wib][0];

  // ---- per-agent weights live in registers for the whole batch loop ----
  float w1a[HID], w1b[HID], b1a[HID], b1b[HID], w2a[HID], w2b[HID];
  float bc1r[HID], wc2r[HID];
#pragma unroll
  for (int h = 0; h < HID; ++h) {
    w1a[h] = W1[n0*HID+h]; w1b[h] = W1[n1*HID+h];
    b1a[h] = b1[n0*HID+h]; b1b[h] = b1[n1*HID+h];
    w2a[h] = W2[n0*HID+h]; w2b[h] = W2[n1*HID+h];
    bc1r[h] = bc1[h];      wc2r[h] = Wc2[h];
  }
  const float b2a  = b2[n0], b2b = b2[n1];
  const float bc2s = bc2[0];

  // WMMA A-operand: Wc1^T padded to 16x4 f32.
  // A layout (16x4 f32): lane m<16 holds row m, VGPR0=K0, VGPR1=K1; lanes 16..31 hold K2,K3 (zero).
  v2f amat;
  amat.x = (lane < 8) ? Wc1[lane]     : 0.0f;   // Wc1[0][h]
  amat.y = (lane < 8) ? Wc1[8 + lane] : 0.0f;   // Wc1[1][h]

  const int  m  = lane & 15;
  const bool lo = lane < 16;

  // uniform (SGPR) row pointers, advanced incrementally: saddr + lane-offset loads
  const size_t stride = (size_t)waves_total * NAG;
  const float* xp = x   + (size_t)gw * NAG;
  float*       op = out + (size_t)gw * NAG;

  for (int b = gw; b < Bsz; b += waves_total, xp += stride, op += stride) {
    const float2 xv = *(const float2*)(xp + n0);        // coalesced 256B/wave
    if (b + waves_total < Bsz)                          // scalar condition
      __builtin_prefetch(xp + stride + n0, 0, 0);

    lx[n0] = xv.x;
    lx[n1] = xv.y;
    __builtin_amdgcn_wave_barrier();   // scheduling fence; DS is in-order per wave

    // ---- coupling MLP: 16 edges per v_wmma_f32_16x16x4_f32 ----
    // D = (Wc1^T) x (feats^T):  D[h][edge] = Wc1[0][h]*x[e] + Wc1[1][h]*x[e+1]
    // -> h sits in the VGPR index, edge in the lane: Wc2-dot becomes per-lane FMAs.
#pragma unroll
    for (int g = 0; g < 4; ++g) {
      v2f bmat;  // B layout (4x16 f32): VGPR0 = K0 row (lanes 0..15) / K2 (zeros), VGPR1 = K1 / K3
      bmat.x = lo ? lx[16*g + m]            : 0.0f;     // x[send] = x[e]
      bmat.y = lo ? lx[(16*g + m + 1) & 63] : 0.0f;     // x[recv] = x[(e+1)%64]
      v8f c = {};
      v8f d = __builtin_amdgcn_wmma_f32_16x16x4_f32(
          false, amat, false, bmat, (short)0, c, false, false);
      float contrib = bc2s;
#pragma unroll
      for (int r = 0; r < HID; ++r)
        contrib += wc2r[r] * fast_tanhf(d[r] + bc1r[r]);
      // unconditional store: upper 16 lanes write to scrap slots -> no EXEC divergence near WMMA
      lc[lo ? (16*g + m) : (NAG + m)] = contrib;
    }
    __builtin_amdgcn_wave_barrier();

    // ---- symmetric scatter (ring): interactions[n] = contrib[n-1] - contrib[n] ----
    const float c_prev = lc[(n0 + 63) & 63];
    const float c0     = lc[n0];
    const float c1     = lc[n1];

    float acc0 = b2a + (c_prev - c0);
    float acc1 = b2b + (c0 - c1);
    // ---- intrinsic per-agent MLP (weights in registers, hardware tanh) ----
#pragma unroll
    for (int h = 0; h < HID; ++h) {
      acc0 += w2a[h] * fast_tanhf(fmaf(xv.x, w1a[h], b1a[h]));
      acc1 += w2b[h] * fast_tanhf(fmaf(xv.y, w1b[h], b1b[h]));
    }
    *(float2*)(op + n0) = make_float2(acc0, acc1);
  }
}

extern "C" void kernel_launch(void* const* d_in, const int* in_sizes, int n_in,
                              void* d_out, int out_size, void* d_ws, size_t ws_size,
                              hipStream_t stream) {
  (void)in_sizes; (void)n_in; (void)d_ws; (void)ws_size;
  const float* x   = (const float*)d_in[0];
  const float* W1  = (const float*)d_in[1];
  const float* b1  = (const float*)d_in[2];
  const float* W2  = (const float*)d_in[3];
  const float* b2  = (const float*)d_in[4];
  const float* Wc1 = (const float*)d_in[5];
  const float* bc1 = (const float*)d_in[6];
  const float* Wc2 = (const float*)d_in[7];
  const float* bc2 = (const float*)d_in[8];
  float* out = (float*)d_out;

  const int Bsz = out_size / NAG;                 // 131072 rows
  int blocks = 2048;                              // 16384 waves -> 8 rows per wave
  int waves_total = blocks * WAVES_PER_BLOCK;
  if (waves_total > Bsz) {
    blocks = (Bsz + WAVES_PER_BLOCK - 1) / WAVES_PER_BLOCK;
    waves_total = blocks * WAVES_PER_BLOCK;
  }
  netode_kernel<<<blocks, BLOCK, 0, stream>>>(x, W1, b1, W2, b2, Wc1, bc1, Wc2, bc2,
                                              out, Bsz, waves_total);
}